// SM2_31585189494808
// MI455X (gfx1250) — compile-verified
//
#include <hip/hip_runtime.h>

// SM-2 scan: seq=512 steps, batch=32768 independent lanes, 3-float state.
// Bandwidth-bound (~335 MB @ 23.3 TB/s ~= 14 us). One lane per batch column,
// single-wave (32-thread) workgroups. Depth-32 async global->LDS prefetch ring
// (8 MB in flight across 1024 waves) + non-temporal output stores.

#define SEQ   512
#define BATCH 32768
#define PFD   32           // prefetch depth (ring slots), power of two, <= 63
#define SMINV 0.01f
#define SMAXV 36500.0f

#if __has_builtin(__builtin_amdgcn_global_load_async_to_lds_b64)
#define USE_ASYNC_LDS 1
typedef int v2i __attribute__((vector_size(8)));
typedef __attribute__((address_space(1))) v2i* gas_v2i_ptr;
typedef __attribute__((address_space(3))) v2i* lds_v2i_ptr;
#else
#define USE_ASYNC_LDS 0
#endif

__device__ __forceinline__ void wait_async_le(int) {}

__device__ __forceinline__ void wait_async_le_pfd1() {
#if __has_builtin(__builtin_amdgcn_s_wait_asynccnt)
  __builtin_amdgcn_s_wait_asynccnt(PFD - 1);
#else
  asm volatile("s_wait_asynccnt %0" :: "n"(PFD - 1));
#endif
}

__device__ __forceinline__ void wait_async_0() {
#if __has_builtin(__builtin_amdgcn_s_wait_asynccnt)
  __builtin_amdgcn_s_wait_asynccnt(0);
#else
  asm volatile("s_wait_asynccnt 0" ::);
#endif
}

__device__ __forceinline__ void wait_ds0() {
#if __has_builtin(__builtin_amdgcn_s_wait_dscnt)
  __builtin_amdgcn_s_wait_dscnt(0);
#else
  asm volatile("s_wait_dscnt 0" ::);
#endif
}

__global__ __launch_bounds__(32)
void SM2_31585189494808_kernel(const float* __restrict__ in,
                               const float* __restrict__ w,
                               float* __restrict__ out) {
  const int lane = threadIdx.x;                 // 0..31 (wave32)
  const int b    = blockIdx.x * 32 + lane;      // batch column

  // w is wave-uniform -> scalar loads
  const float w0 = w[0], w1 = w[1], w2 = w[2];
  const float w3 = w[3], w4 = w[4], w5 = w[5];

  float ivl = 0.0f, ef = w2, reps = 0.0f;

  const float2* __restrict__ gin = (const float2*)in;   // element (t,b) at t*BATCH+b

  // One SM-2 step + non-temporal 12B output store.
  auto step = [&](float2 v, int t) {
    const float rating  = v.y;
    const bool  success = rating > 1.0f;
    reps = success ? (reps + 1.0f) : 1.0f;
    float nivl = (reps == 1.0f) ? w0 : ((reps == 2.0f) ? w1 : ivl * ef);
    const float q  = (rating + 1.0f) - w4;
    float nef = ef - w3 * q * q + w5;
    ivl = fminf(fmaxf(nivl, SMINV), SMAXV);
    ef  = fminf(fmaxf(nef, 1.3f), 10.0f);
    float* p = out + ((size_t)t * BATCH + b) * 3;
    __builtin_nontemporal_store(ivl,  p + 0);
    __builtin_nontemporal_store(ef,   p + 1);
    __builtin_nontemporal_store(reps, p + 2);
  };

#if USE_ASYNC_LDS
  __shared__ float2 stage[PFD * 32];            // 8 KB ring, one wave per WG
  #pragma unroll
  for (int s = 0; s < PFD; ++s) {
    __builtin_amdgcn_global_load_async_to_lds_b64(
        (gas_v2i_ptr)&gin[(size_t)s * BATCH + b],
        (lds_v2i_ptr)&stage[s * 32 + lane],
        0, 0);
  }

  // Steady state: PFD loads always in flight; wait<=PFD-1 retires stage t.
  for (int t = 0; t < SEQ - PFD; ++t) {
    wait_async_le_pfd1();
    const int slot = (t & (PFD - 1)) * 32 + lane;
    float2 v = stage[slot];                     // ds_load_b64
    wait_ds0();                                 // WAR: slot reuse vs async writer
    __builtin_amdgcn_global_load_async_to_lds_b64(
        (gas_v2i_ptr)&gin[(size_t)(t + PFD) * BATCH + b],
        (lds_v2i_ptr)&stage[slot],
        0, 0);
    step(v, t);
  }

  // Drain: all remaining stages have landed after asynccnt==0.
  wait_async_0();
  for (int t = SEQ - PFD; t < SEQ; ++t) {
    float2 v = stage[(t & (PFD - 1)) * 32 + lane];
    step(v, t);
  }
#else
  for (int t = 0; t < SEQ; ++t) {
    float2 v = gin[(size_t)t * BATCH + b];      // global_load_b64
    if (t + PFD < SEQ)
      __builtin_prefetch(&gin[(size_t)(t + PFD) * BATCH + b], 0, 0);
    step(v, t);
  }
#endif

  // final_state appended after outputs (tiny, regular stores)
  float* fp = out + (size_t)SEQ * BATCH * 3 + (size_t)b * 3;
  fp[0] = ivl; fp[1] = ef; fp[2] = reps;
}

extern "C" void kernel_launch(void* const* d_in, const int* in_sizes, int n_in,
                              void* d_out, int out_size, void* d_ws, size_t ws_size,
                              hipStream_t stream) {
  (void)in_sizes; (void)n_in; (void)out_size; (void)d_ws; (void)ws_size;
  const float* in  = (const float*)d_in[0];
  const float* w   = (const float*)d_in[1];
  float*       out = (float*)d_out;
  dim3 grid(BATCH / 32), block(32);
  SM2_31585189494808_kernel<<<grid, block, 0, stream>>>(in, w, out);
}